// TorchEinsumGather_18150531793004
// MI455X (gfx1250) — compile-verified
//
#include <hip/hip_runtime.h>
#include <stdint.h>

// Problem constants (match reference)
constexpr int Bc = 8;     // batch
constexpr int Tc = 8192;  // tokens
constexpr int Ic = 128;   // inner dim (GEMM K)
constexpr int Ec = 16;    // experts
constexpr int Jc = 128;   // output cols (GEMM N)
constexpr int Kc = 1024;  // gathered tokens per (b,e) (GEMM M)

constexpr int MROWS      = 64;   // gathered rows per block (4 x 16-row WMMA M-tiles)
constexpr int LDS_STRIDE = 132;  // floats; pad 128->132 to kill LDS bank conflicts

typedef __attribute__((ext_vector_type(2))) float v2f;
typedef __attribute__((ext_vector_type(8))) float v8f;

// One block per (b, e, 64-row M-chunk). 256 threads = 8 wave32.
// Wave w owns output columns [16w, 16w+16): it preloads its 128x16 B slice of
// W[e] into 64 VGPRs ONCE and reuses it across four 16x16 M-tiles (128 WMMAs).
__global__ __launch_bounds__(256)
void gather_gemm_wmma_f32(const float* __restrict__ X,      // [B,T,I]
                          const int64_t* __restrict__ ind,  // [B,E,K]
                          const float* __restrict__ W,      // [E,I,J]
                          float* __restrict__ Y)            // [B,E,K,J]
{
    __shared__ float xs[MROWS * LDS_STRIDE];  // gathered A tile, 64 x 128 (padded)

    const int kt = blockIdx.x;  // 0..15   64-row M-chunk index
    const int e  = blockIdx.y;  // 0..15
    const int b  = blockIdx.z;  // 0..7

    const int tid = threadIdx.x;

    // ---- Cooperative gather: 64 indexed rows of X[b] -> LDS ----
    // 16 threads per row, each thread moves 8 floats (2x float4), 4 row-waves.
    {
        const int lane16 = tid & 15;
        const int rbase  = tid >> 4;  // 0..15
        const int c0     = lane16 * 8;
        const int64_t ibase = ((int64_t)(b * Ec + e)) * Kc + (int64_t)kt * MROWS;
#pragma unroll
        for (int mt = 0; mt < 4; ++mt) {
            const int row = mt * 16 + rbase;
            const int64_t tok = ind[ibase + row];
            const float* src = X + ((int64_t)b * Tc + tok) * Ic;
            float* dstl = &xs[row * LDS_STRIDE];
            *(float4*)(dstl + c0)     = *(const float4*)(src + c0);
            *(float4*)(dstl + c0 + 4) = *(const float4*)(src + c0 + 4);
        }
    }
    __syncthreads();

    const int wave  = tid >> 5;   // 0..7  -> N-tile
    const int lane  = tid & 31;
    const int ln    = lane & 15;  // N (for B/C) or M (for A)
    const int khalf = lane >> 4;  // 0/1: which K-pair this half-wave holds

    // ---- Preload B: W[e][:, 16w .. 16w+15], 128x16 f32 -> 64 VGPRs ----
    // v_wmma_f32_16x16x4_f32 B layout: VGPR0 = {K=0 | K=2}, VGPR1 = {K=1 | K=3}
    // lanes 0-15 carry N=0..15 (contiguous j -> coalesced 64B loads).
    const float* Wb = W + (int64_t)e * Ic * Jc + wave * 16 + ln;
    v2f breg[32];
#pragma unroll
    for (int kk = 0; kk < 32; ++kk) {
        const int i0 = kk * 4 + khalf * 2;
        v2f bv;
        bv.x = Wb[(int64_t)i0 * Jc];
        bv.y = Wb[(int64_t)(i0 + 1) * Jc];
        breg[kk] = bv;
    }

    // ---- 4 M-tiles x 32 V_WMMA_F32_16X16X4_F32, B held in registers ----
    // A layout: lanes carry M=0..15; VGPR0 = {K=0 | K=2}, VGPR1 = {K=1 | K=3}
    // -> lane reads xs[m][4kk + 2*khalf .. +1] (8B LDS load, conflict-free:
    //    bank = (4m + i) mod 64 is distinct across the 16 lanes of each half).
    float* const dst0 = Y + (((int64_t)(b * Ec + e)) * Kc + (int64_t)kt * MROWS) * Jc
                          + wave * 16 + ln;
#pragma unroll
    for (int mt = 0; mt < 4; ++mt) {
        const float* arow = &xs[(mt * 16 + ln) * LDS_STRIDE + khalf * 2];
        v8f c = {};
#pragma unroll
        for (int kk = 0; kk < 32; ++kk) {
            v2f a;
            a.x = arow[kk * 4];
            a.y = arow[kk * 4 + 1];
            // 8 args: (neg_a, A, neg_b, B, c_mod, C, reuse_a, reuse_b)
            c = __builtin_amdgcn_wmma_f32_16x16x4_f32(
                    false, a, false, breg[kk], (short)0, c, false, false);
        }
        // ---- Store C non-temporally (64MB pure write stream; keep X in L2) ----
        // lane L, VGPR v -> row m = v + 8*khalf, col n = ln.
        float* dst = dst0 + (int64_t)(mt * 16) * Jc;
#pragma unroll
        for (int v = 0; v < 8; ++v) {
            const int m = v + khalf * 8;
            __builtin_nontemporal_store(c[v], dst + (int64_t)m * Jc);
        }
    }
}

extern "C" void kernel_launch(void* const* d_in, const int* in_sizes, int n_in,
                              void* d_out, int out_size, void* d_ws, size_t ws_size,
                              hipStream_t stream) {
    const float*   X   = (const float*)d_in[0];
    const int64_t* ind = (const int64_t*)d_in[1];
    const float*   W   = (const float*)d_in[2];
    float*         Y   = (float*)d_out;

    dim3 grid(Kc / MROWS, Ec, Bc);  // (16, 16, 8)
    gather_gemm_wmma_f32<<<grid, 256, 0, stream>>>(X, ind, W, Y);
}